// Mypnufft_liverMRFMGRE_3D_v2_48704929136934
// MI455X (gfx1250) — compile-verified
//
#include <hip/hip_runtime.h>
#include <math.h>

typedef __attribute__((ext_vector_type(2))) float v2f;
typedef __attribute__((ext_vector_type(8))) float v8f;

#define IMG    64
#define NSLICE 16
#define NCOIL  4
#define NSING  4
#define NECHO  3
#define NRO    32
#define NPT    96
#define S_TOT  (NSING * NECHO)   // 12
#define PT_TILES (NPT / 16)      // 6

// ---- workspace layout (floats) ----
// ex : [r][x/2][p][2]  (x-pair innermost -> one b64 per A fragment, p lane-contig)
// ey : [r][p][y]       (y fastest -> post-WMMA loads lane-contiguous)
// scf: [s][c][z][x/2][y][2]  (x-pair innermost -> B fragment = 2 adjacent words)
#define EX_SZ     (NRO * IMG * NPT)                    // 196608
#define SCF_SZ    (S_TOT * NCOIL * NSLICE * IMG * IMG) // 3145728
#define EXRE_OFF  0
#define EXIM_OFF  (EXRE_OFF + EX_SZ)
#define EYRE_OFF  (EXIM_OFF + EX_SZ)
#define EYIM_OFF  (EYRE_OFF + EX_SZ)
#define SCFRE_OFF (EYIM_OFF + EX_SZ)
#define SCFIM_OFF (SCFRE_OFF + SCF_SZ)

#define LDS_ROW 129           // 128 payload + 1 pad (odd stride -> half-waves hit
                              // disjoint bank parities on B-frag ds_load_b32)

// ---------------- Kernel A: phase tables  ex = exp(-i*w0*xs), ey = exp(-i*w1*ys)
__global__ void build_phase(const float* __restrict__ ktraj, float* __restrict__ ws) {
    int t = blockIdx.x * blockDim.x + threadIdx.x;
    const int total = NRO * NPT * IMG;
    if (t >= total) return;
    int i = t % IMG;
    int p = (t / IMG) % NPT;
    int r = t / (IMG * NPT);
    float w0 = ktraj[(r * NPT + p) * 2 + 0];
    float w1 = ktraj[(r * NPT + p) * 2 + 1];
    float coord = (float)(i - IMG / 2);
    float s0, c0, s1, c1;
    __sincosf(w0 * coord, &s0, &c0);
    __sincosf(w1 * coord, &s1, &c1);
    // paired-x layout: [(r*32 + i/2)*96 + p]*2 + (i&1)
    long exo = ((long)(r * (IMG / 2) + (i >> 1)) * NPT + p) * 2 + (i & 1);
    ws[EXRE_OFF + exo] = c0;
    ws[EXIM_OFF + exo] = -s0;                    // exp(-i a) = cos a - i sin a
    ws[EYRE_OFF + (r * NPT + p) * IMG + i] = c1;
    ws[EYIM_OFF + (r * NPT + p) * IMG + i] = -s1;
}

// ---------------- Kernel B: scf = FFT_z( singulars * (-1)^z * smap )
__global__ void build_scf(const float* __restrict__ sing,
                          const float* __restrict__ smap,
                          float* __restrict__ ws) {
    int t = blockIdx.x * blockDim.x + threadIdx.x;  // sites: s,c,x,y
    const int total = S_TOT * NCOIL * IMG * IMG;
    if (t >= total) return;
    int y = t % IMG;
    int x = (t / IMG) % IMG;
    int c = (t / (IMG * IMG)) % NCOIL;
    int s = t / (IMG * IMG * NCOIL);
    int n = s / NECHO, e = s % NECHO;

    float ar[NSLICE], ai[NSLICE];
#pragma unroll
    for (int z = 0; z < NSLICE; ++z) {
        long si = ((((long)x * IMG + y) * NSLICE + z) * NSING + n) * NECHO + e;
        float sr = sing[si * 2 + 0], sii = sing[si * 2 + 1];
        float sign = (z & 1) ? -1.0f : 1.0f;
        sr *= sign; sii *= sign;
        long mi = (((long)c * IMG + x) * IMG + y) * NSLICE + z;
        float mr = smap[mi * 2 + 0], mii = smap[mi * 2 + 1];
        ar[z] = sr * mr - sii * mii;
        ai[z] = sr * mii + sii * mr;
    }
    float twc[NSLICE], tws[NSLICE];
#pragma unroll
    for (int m = 0; m < NSLICE; ++m) {
        float ang = -(float)(M_PI / 8.0) * (float)m;
        __sincosf(ang, &tws[m], &twc[m]);
    }
    // pair-x element offset within a z-plane
    long eo = ((long)(x >> 1) * IMG + y) * 2 + (x & 1);
#pragma unroll
    for (int k = 0; k < NSLICE; ++k) {
        float orr = 0.0f, oii = 0.0f;
#pragma unroll
        for (int z = 0; z < NSLICE; ++z) {
            const int m = (k * z) & (NSLICE - 1);   // constant after unroll
            orr += ar[z] * twc[m] - ai[z] * tws[m];
            oii += ar[z] * tws[m] + ai[z] * twc[m];
        }
        long o = ((long)(s * NCOIL + c) * NSLICE + k) * (IMG * IMG) + eo;
        ws[SCFRE_OFF + o] = orr;
        ws[SCFIM_OFF + o] = oii;
    }
}

// ---------------- Kernel C: zero output (harness poisons d_out)
__global__ void zero_out(float* __restrict__ out, int n) {
    int t = blockIdx.x * blockDim.x + threadIdx.x;
    if (t < n) out[t] = 0.0f;
}

// ---------------- Kernel D: complex NUFFT GEMM via V_WMMA_F32_16X16X4_F32
// One block = 6 waves = all p-tiles of one (r,s,c); B slab staged once in LDS.
__global__ __launch_bounds__(32 * PT_TILES)
void nufft_wmma(const float* __restrict__ ws,
                const int* __restrict__ sbin,
                const float* __restrict__ Ur,
                float* __restrict__ out) {
    __shared__ float ldsRe[(IMG / 2) * LDS_ROW];   // 32 x2-rows of 129 floats
    __shared__ float ldsIm[(IMG / 2) * LDS_ROW];

    int b = blockIdx.x;                  // NRO * S_TOT * NCOIL
    int c = b % NCOIL;
    int s = (b / NCOIL) % S_TOT;
    int r = b / (NCOIL * S_TOT);
    int z = sbin[r];
    int n = s / NECHO, e = s % NECHO;

    // ---- stage the [64 x 64] complex scf slab into LDS (32 KB) ----
    long plane = ((long)(s * NCOIL + c) * NSLICE + z) * (IMG * IMG);
    const float4* gRe = (const float4*)(ws + SCFRE_OFF + plane);
    const float4* gIm = (const float4*)(ws + SCFIM_OFF + plane);
    for (int idx = threadIdx.x; idx < 2048; idx += 32 * PT_TILES) {
        int pl   = idx >> 10;            // 0 = re, 1 = im
        int f    = idx & 1023;           // float4 chunk within plane
        int row  = f >> 5;               // x2-row (128 floats = 32 float4)
        int col4 = f & 31;
        float4 v = pl ? gIm[f] : gRe[f];
        float* dst = (pl ? ldsIm : ldsRe) + row * LDS_ROW + col4 * 4;
        dst[0] = v.x; dst[1] = v.y; dst[2] = v.z; dst[3] = v.w;
    }
    __syncthreads();

    int wave = threadIdx.x >> 5;         // = p-tile index
    int lane = threadIdx.x & 31;
    int lh = lane & 15;                  // frag row/col within half
    int hi = lane >> 4;                  // K' = j + 2*hi ; rows m = 8*hi + j
    const int p0 = wave * 16;

    // ---- hoist ALL A fragments (yt-invariant): 16 K-steps x (re,im) ----
    const v2f* exReP = (const v2f*)(ws + EXRE_OFF) + (long)r * (IMG / 2) * NPT;
    const v2f* exImP = (const v2f*)(ws + EXIM_OFF) + (long)r * (IMG / 2) * NPT;
    v2f arA[16], aiA[16];
#pragma unroll
    for (int kk = 0; kk < 16; ++kk) {
        int x2 = 2 * kk + hi;            // pair row; pair = (K=j+2hi for j=0,1)
        arA[kk] = exReP[x2 * NPT + p0 + lh];   // one b64, lane-contiguous
        aiA[kk] = exImP[x2 * NPT + p0 + lh];
    }

    const float* eyRe = ws + EYRE_OFF + (long)r * NPT * IMG;
    const float* eyIm = ws + EYIM_OFF + (long)r * NPT * IMG;

    float nufRe[8], nufIm[8];
#pragma unroll
    for (int j = 0; j < 8; ++j) { nufRe[j] = 0.0f; nufIm[j] = 0.0f; }

    for (int yt = 0; yt < IMG / 16; ++yt) {
        v8f accRe = {};
        v8f accIm = {};
        const int y = yt * 16 + lh;
        const int yoff = y * 2;
#pragma unroll
        for (int kk = 0; kk < 16; ++kk) {
            int rowbase = (2 * kk + hi) * LDS_ROW + yoff;
            v2f brf, bif, nbif;
            brf[0] = ldsRe[rowbase];     bif[0] = ldsIm[rowbase];
            brf[1] = ldsRe[rowbase + 1]; bif[1] = ldsIm[rowbase + 1];
            nbif = -bif;                 // f32 WMMA has no A/B negation
            // complex MAC: Re += ar*br - ai*bi ; Im += ar*bi + ai*br
            accRe = __builtin_amdgcn_wmma_f32_16x16x4_f32(false, arA[kk], false, brf,  (short)0, accRe, false, false);
            accRe = __builtin_amdgcn_wmma_f32_16x16x4_f32(false, aiA[kk], false, nbif, (short)0, accRe, false, false);
            accIm = __builtin_amdgcn_wmma_f32_16x16x4_f32(false, arA[kk], false, bif,  (short)0, accIm, false, false);
            accIm = __builtin_amdgcn_wmma_f32_16x16x4_f32(false, aiA[kk], false, brf,  (short)0, accIm, false, false);
        }
        // fold in ey[p,y]: lane holds rows m = 8*hi + j, col y
#pragma unroll
        for (int j = 0; j < 8; ++j) {
            int p = p0 + 8 * hi + j;
            float er = eyRe[p * IMG + y];
            float ei = eyIm[p * IMG + y];
            nufRe[j] += accRe[j] * er - accIm[j] * ei;
            nufIm[j] += accRe[j] * ei + accIm[j] * er;
        }
    }

    // reduce over the 16 lanes (y) within each half-wave
#pragma unroll
    for (int j = 0; j < 8; ++j) {
        float vr = nufRe[j], vi = nufIm[j];
        for (int m = 1; m <= 8; m <<= 1) {
            vr += __shfl_xor(vr, m, 32);
            vi += __shfl_xor(vi, m, 32);
        }
        nufRe[j] = vr; nufIm[j] = vi;
    }

    if (lh == 0) {
        float u = Ur[r * NSING + n];     // Ur_list[0, r, n]
#pragma unroll
        for (int j = 0; j < 8; ++j) {
            int p = p0 + 8 * hi + j;
            long o = (((long)(p * NCOIL + c) * NRO + r) * NECHO + e) * 2;
            atomicAdd(&out[o + 0], u * nufRe[j]);   // n-sum collides -> atomic
            atomicAdd(&out[o + 1], u * nufIm[j]);
        }
    }
}

extern "C" void kernel_launch(void* const* d_in, const int* in_sizes, int n_in,
                              void* d_out, int out_size, void* d_ws, size_t ws_size,
                              hipStream_t stream) {
    const float* singulars = (const float*)d_in[0];  // [64,64,16,4,3,2]
    const float* smap      = (const float*)d_in[1];  // [4,64,64,16,2]
    const float* ktraj     = (const float*)d_in[2];  // [1,32,96,2]
    const float* Ur        = (const float*)d_in[3];  // [1,32,4]
    const int*   sbin      = (const int*)  d_in[4];  // [32]
    (void)in_sizes; (void)n_in; (void)ws_size;       // dc (d_in[5]) unused by fwd
    float* out = (float*)d_out;
    float* ws  = (float*)d_ws;

    {   // phase tables
        int total = NRO * NPT * IMG;
        build_phase<<<(total + 255) / 256, 256, 0, stream>>>(ktraj, ws);
    }
    {   // coil-combined, FFT'd slice images
        int total = S_TOT * NCOIL * IMG * IMG;
        build_scf<<<(total + 255) / 256, 256, 0, stream>>>(singulars, smap, ws);
    }
    zero_out<<<(out_size + 255) / 256, 256, 0, stream>>>(out, out_size);

    int blocks = NRO * S_TOT * NCOIL;                // 1536 blocks x 6 waves
    nufft_wmma<<<blocks, 32 * PT_TILES, 0, stream>>>(ws, sbin, Ur, out);
}